// Channel2DTransformer_16398185136177
// MI455X (gfx1250) — compile-verified
//
#include <hip/hip_runtime.h>

typedef __attribute__((ext_vector_type(2))) float v2f;
typedef __attribute__((ext_vector_type(8))) float v8f;

#define N_HEADS 32                    // the 'a'/'f' dimension
#define C_DIM   32
#define B_DIM   4
#define HW      4096                  // 64*64
#define ROWSTRIDE (C_DIM * HW)        // x stride between consecutive n
#define OUT_A_STRIDE (B_DIM * C_DIM * HW)

// ---------------------------------------------------------------------------
// Kernel A: partial Gram matrices  G_part = X_chunk * X_chunk^T
// grid = 128 jobs * 4 K-splits, block = 256 (8 wave32)
// Each wave covers 128 K-columns, accumulating a 32x32 tile grid (2x2 WMMA
// tiles) with V_WMMA_F32_16X16X4_F32. Deterministic tree reduction in LDS.
// ---------------------------------------------------------------------------
__global__ __launch_bounds__(256) void gram_kernel(const float* __restrict__ x,
                                                   float* __restrict__ Gpart) {
  __shared__ float Gw[8 * 1024];      // per-wave 32x32 partials
  const int bc     = blockIdx.x >> 2;
  const int ksplit = blockIdx.x & 3;
  const int b = bc >> 5, c = bc & 31;
  const int tid = threadIdx.x;
  const int wave = tid >> 5, lane = tid & 31;
  const int r  = lane & 15;           // matrix row within 16-row group
  const int kh = (lane >> 4) << 1;    // K sub-offset: 0 or 2

  const size_t xbase = ((size_t)(b * N_HEADS) * C_DIM + c) * HW;
  const float* row0 = x + xbase + (size_t)r * ROWSTRIDE;        // rows 0..15
  const float* row1 = row0 + (size_t)16 * ROWSTRIDE;            // rows 16..31

  v8f a00 = {}, a01 = {}, a10 = {}, a11 = {};
  const int klo = ksplit * 1024 + wave * 128;
  #pragma unroll 4
  for (int k0 = klo; k0 < klo + 128; k0 += 4) {
    v2f op0 = *(const v2f*)(row0 + k0 + kh);   // serves as both A and B operand
    v2f op1 = *(const v2f*)(row1 + k0 + kh);
    a00 = __builtin_amdgcn_wmma_f32_16x16x4_f32(false, op0, false, op0, (short)0, a00, false, false);
    a01 = __builtin_amdgcn_wmma_f32_16x16x4_f32(false, op0, false, op1, (short)0, a01, false, false);
    a10 = __builtin_amdgcn_wmma_f32_16x16x4_f32(false, op1, false, op0, (short)0, a10, false, false);
    a11 = __builtin_amdgcn_wmma_f32_16x16x4_f32(false, op1, false, op1, (short)0, a11, false, false);
  }

  // D layout: VGPR j, lane L -> (M = j + 8*(L>>4), N = L&15)
  const int nn = lane & 15;
  const int mh = (lane >> 4) << 3;
  float* gw = Gw + wave * 1024;
  #pragma unroll
  for (int j = 0; j < 8; ++j) {
    gw[(mh + j) * 32 + nn]           = a00[j];
    gw[(mh + j) * 32 + 16 + nn]      = a01[j];
    gw[(16 + mh + j) * 32 + nn]      = a10[j];
    gw[(16 + mh + j) * 32 + 16 + nn] = a11[j];
  }
  __syncthreads();

  for (int i = tid; i < 1024; i += 256) {
    float s = 0.f;
    #pragma unroll
    for (int wv = 0; wv < 8; ++wv) s += Gw[wv * 1024 + i];
    Gpart[(size_t)blockIdx.x * 1024 + i] = s;
  }
}

// ---------------------------------------------------------------------------
// Kernel B: sum the 4 K-split partials, apply per-channel weights + scale,
// row-softmax, fold wv in:  Ap[a][f] = softmax_f((1/64)*wq[a]*wk[f]*G) * wv[f]
// grid = 128, block = 32 (one thread per row a)
// ---------------------------------------------------------------------------
__global__ __launch_bounds__(32) void softmax_kernel(const float* __restrict__ Gpart,
                                                     const float* __restrict__ w,
                                                     float* __restrict__ Ap) {
  __shared__ float sWk[32], sWv[32];
  const int bc = blockIdx.x;
  const int c  = bc & 31;
  const int a  = threadIdx.x;
  const int wo = (a * C_DIM + c) * 3;     // depthwise 1x1 weight index
  const float wqa = w[wo + 0];
  sWk[a] = w[wo + 1];
  sWv[a] = w[wo + 2];
  __syncthreads();

  const float* gp = Gpart + (size_t)bc * 4096;   // [4 splits][1024]
  float srow[32];
  float mx = -3.4e38f;
  const float sc = 0.015625f * wqa;              // (H*W)^-0.5 = 1/64
  #pragma unroll
  for (int f = 0; f < 32; ++f) {
    const int i = a * 32 + f;
    const float g = gp[i] + gp[1024 + i] + gp[2048 + i] + gp[3072 + i];
    const float s = sc * sWk[f] * g;
    srow[f] = s;
    mx = fmaxf(mx, s);
  }
  float sum = 0.f;
  #pragma unroll
  for (int f = 0; f < 32; ++f) { const float e = expf(srow[f] - mx); srow[f] = e; sum += e; }
  const float inv = 1.f / sum;
  float* ap = Ap + (size_t)bc * 1024 + a * 32;
  #pragma unroll
  for (int f = 0; f < 32; ++f) ap[f] = srow[f] * inv * sWv[f];
}

// ---------------------------------------------------------------------------
// Kernel C: out[a, b, c, hw] = sum_f Ap[a][f] * X[f][hw]   (32x32 x 32x4096)
// grid = 128 jobs * 8 HW-splits, block = 256 (8 waves); each wave owns 64 cols.
// X comes out of L2 (whole x = 64 MiB << 192 MB L2).
// ---------------------------------------------------------------------------
__global__ __launch_bounds__(256) void outmm_kernel(const float* __restrict__ x,
                                                    const float* __restrict__ Ap,
                                                    float* __restrict__ out) {
  const int bc      = blockIdx.x >> 3;
  const int hwsplit = blockIdx.x & 7;
  const int b = bc >> 5, c = bc & 31;
  const int tid = threadIdx.x;
  const int wave = tid >> 5, lane = tid & 31;
  const int r  = lane & 15;
  const int kh = (lane >> 4) << 1;      // K sub-offset 0/2
  const int nn = lane & 15;
  const int mh = (lane >> 4) << 3;

  const size_t xbase = ((size_t)(b * N_HEADS) * C_DIM + c) * HW;
  const float* ap = Ap + (size_t)bc * 1024;

  // Preload all A-operand fragments (A' is only 32x32).
  v2f Aop[2][8];
  #pragma unroll
  for (int tm = 0; tm < 2; ++tm)
    #pragma unroll
    for (int kk = 0; kk < 8; ++kk)
      Aop[tm][kk] = *(const v2f*)(ap + (tm * 16 + r) * 32 + kk * 4 + kh);

  float* outb = out + ((size_t)b * C_DIM + c) * HW;   // out[a][b][c][hw]

  const int hwbase = hwsplit * 512 + wave * 64;
  #pragma unroll
  for (int t = 0; t < 4; ++t) {
    const int hw0 = hwbase + t * 16;
    v8f d0 = {}, d1 = {};
    #pragma unroll
    for (int kk = 0; kk < 8; ++kk) {
      // B 4x16 fragment: lane holds (K=kk*4+kh, N=nn) and (K=kk*4+kh+1, N=nn)
      const float* bp = x + xbase + (size_t)(kk * 4 + kh) * ROWSTRIDE + hw0 + nn;
      v2f Bop;
      Bop.x = bp[0];
      Bop.y = bp[ROWSTRIDE];
      d0 = __builtin_amdgcn_wmma_f32_16x16x4_f32(false, Aop[0][kk], false, Bop, (short)0, d0, false, false);
      d1 = __builtin_amdgcn_wmma_f32_16x16x4_f32(false, Aop[1][kk], false, Bop, (short)0, d1, false, false);
    }
    #pragma unroll
    for (int j = 0; j < 8; ++j) {
      outb[(size_t)(mh + j) * OUT_A_STRIDE + hw0 + nn]      = d0[j];
      outb[(size_t)(16 + mh + j) * OUT_A_STRIDE + hw0 + nn] = d1[j];
    }
  }
}

extern "C" void kernel_launch(void* const* d_in, const int* in_sizes, int n_in,
                              void* d_out, int out_size, void* d_ws, size_t ws_size,
                              hipStream_t stream) {
  (void)in_sizes; (void)n_in; (void)out_size; (void)ws_size;
  const float* x = (const float*)d_in[0];
  const float* w = (const float*)d_in[1];
  float* out = (float*)d_out;

  float* Gpart = (float*)d_ws;                   // 128*4*1024 floats = 2 MiB
  float* Ap    = Gpart + 128 * 4 * 1024;         // 128*1024 floats = 512 KiB

  gram_kernel   <<<dim3(128 * 4), dim3(256), 0, stream>>>(x, Gpart);
  softmax_kernel<<<dim3(128),     dim3(32),  0, stream>>>(Gpart, w, Ap);
  outmm_kernel  <<<dim3(128 * 8), dim3(256), 0, stream>>>(x, Ap, out);
}